// DepthWiseIIRConv_82549271429627
// MI455X (gfx1250) — compile-verified
//
#include <hip/hip_runtime.h>

// DepthWiseIIRConv: B=256, C=512, H=W=16, fp32.
// One fused pass: depthwise 3x3 conv + 2D causal IIR scan + bias.
// Memory-bound (268 MB @ 23.3 TB/s ~= 11.5 us floor): each byte of x read
// once (coalesced b32 per row across the wave), all cross-lane traffic done
// with DPP16 row shifts in the VALU (no LDS, no waits), bias added exactly
// via V_WMMA_F32_16X16X4_F32.

#define B_ 256
#define C_ 512
#define H_ 16
#define W_ 16

// DPP16 controls: shift within each row of 16 lanes (= one image row here).
#define ROW_SHR(n) (0x110 | (n))  // lane i <- lane i-n (invalid -> old)
#define ROW_SHL(n) (0x100 | (n))  // lane i <- lane i+n (invalid -> old)

// Shift with zero fill: old=0 so out-of-row lanes read 0.0f regardless of
// bound_ctrl interpretation. Pure VALU, no wait counters.
template <int CTRL>
__device__ __forceinline__ float dpp_zf(float v) {
  return __int_as_float(__builtin_amdgcn_update_dpp(
      0, __float_as_int(v), CTRL, 0xF, 0xF, false));
}

typedef __attribute__((ext_vector_type(2))) float v2f;
typedef __attribute__((ext_vector_type(8))) float v8f;

__global__ __launch_bounds__(256) void diir_fused_kernel(
    const float* __restrict__ x, const float* __restrict__ wa,
    const float* __restrict__ bias, float* __restrict__ out) {
  const int lane = threadIdx.x & 31;
  const int waveInBlk = threadIdx.x >> 5;
  const int pair = blockIdx.x * 8 + waveInBlk;   // [0, B*C/2)
  const int ch = pair % C_;                      // channel (wave-uniform)
  const int b = (pair / C_) * 2 + (lane >> 4);   // lanes 0-15: b0, 16-31: b0+1
  const int w = lane & 15;                       // column within plane
  const size_t base = ((size_t)b * C_ + ch) * (H_ * W_);
  const float* xp = x + base;
  float* op = out + base;

  // Per-channel 3x3 weights (cross-correlation, matches reference) + bias.
  float k[9];
#pragma unroll
  for (int i = 0; i < 9; ++i) k[i] = wa[ch * 9 + i];
  const float bv = bias[ch];

  // Load the full plane: lane holds column w over all 16 rows.
  // Per row h the wave reads two contiguous 64B rows -> fully coalesced.
  float xc[16], xl[16], xr[16];
#pragma unroll
  for (int h = 0; h < 16; ++h)
    xc[h] = __builtin_nontemporal_load(xp + h * 16 + w);

  // Horizontal neighbors via DPP16 row shifts; zero-filled at plane edges.
#pragma unroll
  for (int h = 0; h < 16; ++h) {
    xl[h] = dpp_zf<ROW_SHR(1)>(xc[h]);  // x[h][w-1], 0 at w==0
    xr[h] = dpp_zf<ROW_SHL(1)>(xc[h]);  // x[h][w+1], 0 at w==15
  }

  // Depthwise 3x3: ax[h][w] = sum_{dh,dw} x[h-1+dh][w-1+dw] * k[dh*3+dw].
  float ax[16];
#pragma unroll
  for (int h = 0; h < 16; ++h) {
    float s = k[3] * xl[h] + k[4] * xc[h] + k[5] * xr[h];
    if (h > 0)  s += k[0] * xl[h - 1] + k[1] * xc[h - 1] + k[2] * xr[h - 1];
    if (h < 15) s += k[6] * xl[h + 1] + k[7] * xc[h + 1] + k[8] * xr[h + 1];
    ax[h] = s;
  }

  // 2D causal IIR: out_h = cumsum_w( shift(prev) + prev + ax_h ).
  // Sequential over 16 rows; 16-wide inclusive scan = 4 DPP shift-adds.
  float p = 0.0f;
#pragma unroll
  for (int h = 0; h < 16; ++h) {
    float t = dpp_zf<ROW_SHR(1)>(p) + p + ax[h];
    t += dpp_zf<ROW_SHR(1)>(t);
    t += dpp_zf<ROW_SHR(2)>(t);
    t += dpp_zf<ROW_SHR(4)>(t);
    t += dpp_zf<ROW_SHR(8)>(t);
    p = t;
    ax[h] = t;
  }

#if __has_builtin(__builtin_amdgcn_wmma_f32_16x16x4_f32)
  // Exact bias add via D = A x B + C (f32 WMMA, K=4).
  // C: rows 0-7 of both planes are already in the documented C/D layout
  //    (lane = N = w; VGPR r holds M=r for lanes 0-15, M=r+8 for 16-31).
  // A: ones-column in documented A layout -> A[M][0] = 1, all else 0.
  // B: bias replicated in every K slot -> D = 1*bias + C element-wise,
  //    robust to any B-operand K/N striping.
  v2f Am;
  Am.x = (lane < 16) ? 1.0f : 0.0f;  // VGPR0: K=0 (lanes 0-15), K=2 (16-31)
  Am.y = 0.0f;                       // VGPR1: K=1 / K=3
  v2f Bm;
  Bm.x = bv;
  Bm.y = bv;
  v8f c0, c1;
#pragma unroll
  for (int r = 0; r < 8; ++r) { c0[r] = ax[r]; c1[r] = ax[r + 8]; }
  c0 = __builtin_amdgcn_wmma_f32_16x16x4_f32(false, Am, false, Bm, (short)0,
                                             c0, false, false);
  c1 = __builtin_amdgcn_wmma_f32_16x16x4_f32(false, Am, false, Bm, (short)0,
                                             c1, false, false);
#pragma unroll
  for (int r = 0; r < 8; ++r) {
    __builtin_nontemporal_store(c0[r], op + r * 16 + w);
    __builtin_nontemporal_store(c1[r], op + (r + 8) * 16 + w);
  }
#else
#pragma unroll
  for (int h = 0; h < 16; ++h)
    __builtin_nontemporal_store(ax[h] + bv, op + h * 16 + w);
#endif
}

extern "C" void kernel_launch(void* const* d_in, const int* in_sizes, int n_in,
                              void* d_out, int out_size, void* d_ws,
                              size_t ws_size, hipStream_t stream) {
  (void)in_sizes; (void)n_in; (void)out_size; (void)d_ws; (void)ws_size;
  const float* x = (const float*)d_in[0];
  const float* wa = (const float*)d_in[1];     // (1, C, 3, 3)
  const float* bias = (const float*)d_in[2];   // (1, C, 1, 1)
  float* out = (float*)d_out;
  const int pairs = (B_ * C_) / 2;  // 65536 plane-pairs, one per wave
  const int blocks = pairs / 8;     // 8 waves per 256-thread block -> 8192
  diir_fused_kernel<<<blocks, 256, 0, stream>>>(x, wa, bias, out);
}